// HilbertHuangTransform_81630148428275
// MI455X (gfx1250) — compile-verified
//
#include <hip/hip_runtime.h>
#include <math.h>

// ---------------- problem constants ----------------
#define B_   8
#define T_   4096
#define F_   128
#define K_   5          // MAX_IMFS
#define NS_  4          // sift iterations
#define HW_  12         // half of W_ENV=25
#define SSEG 25         // van Herk segment size == window size
#define NSEG ((T_ + SSEG - 1) / SSEG)   // 164
#define PI_F     3.14159265358979323846f
#define TWO_PI_F 6.28318530717958647692f

typedef float v2f __attribute__((ext_vector_type(2)));
typedef float v8f __attribute__((ext_vector_type(8)));

// D = A(16x4) * B(4x16) + C ; f32 WMMA (codegen-verified round 1)
#define WMMA4(Av, Bv, Cv) \
  __builtin_amdgcn_wmma_f32_16x16x4_f32(false, (Av), false, (Bv), (short)0, (Cv), false, false)

#define USE_ASYNC 1

// per-lane async gather: copies 4 bytes from the global address g to the LDS address lds  (ASYNCcnt-tracked)
__device__ __forceinline__ void async_ld32(const float* g, float* lds) {
#if USE_ASYNC
  unsigned l = (unsigned)(size_t)lds;
  unsigned long long ga = (unsigned long long)(size_t)g;
  asm volatile("global_load_async_to_lds_b32 %0, %1, off" :: "v"(l), "v"(ga) : "memory");
#else
  *lds = *g;
#endif
}
__device__ __forceinline__ void async_ld128(const float* g, float* lds) {
#if USE_ASYNC
  unsigned l = (unsigned)(size_t)lds;
  unsigned long long ga = (unsigned long long)(size_t)g;
  asm volatile("global_load_async_to_lds_b128 %0, %1, off" :: "v"(l), "v"(ga) : "memory");
#else
  lds[0] = g[0]; lds[1] = g[1]; lds[2] = g[2]; lds[3] = g[3];
#endif
}
// per-lane async scatter: copies 4 bytes from the LDS address lds to the global address g
__device__ __forceinline__ void async_st32(float* g, const float* lds) {
#if USE_ASYNC
  unsigned l = (unsigned)(size_t)lds;
  unsigned long long ga = (unsigned long long)(size_t)g;
  asm volatile("global_store_async_from_lds_b32 %1, %0, off" :: "v"(l), "v"(ga) : "memory");
#else
  *g = *lds;
#endif
}
__device__ __forceinline__ void async_wait_all() {
#if USE_ASYNC
  asm volatile("s_wait_asynccnt 0x0" ::: "memory");
#endif
}

// ---------------- small param kernel ----------------
__global__ void k_params(const float* __restrict__ cs, const float* __restrict__ fb,
                         int* __restrict__ target, int* __restrict__ boundary) {
  int f = threadIdx.x;
  if (f >= F_) return;
  float c = cs[f];
  float sc = 1.f / (1.f + __expf(-10.f * (c - 0.5f)));
  int tg = 2 + (int)floorf(sc * (float)(K_ - 2));
  tg = tg < 2 ? 2 : (tg > K_ ? K_ : tg);
  target[f] = tg;
  int idx = 0;
#pragma unroll
  for (int j = 0; j < K_ - 1; ++j) idx += (fb[j] < c) ? 1 : 0;   // searchsorted (left)
  int bd = idx + 1;
  bd = bd < 1 ? 1 : (bd > K_ - 1 ? K_ - 1 : bd);
  boundary[f] = bd;
}

// ---------------- transpose x[B,T,F] -> r[B,F,T] (async tile loads) ----------------
__global__ __launch_bounds__(256) void k_transpose(const float* __restrict__ x,
                                                   float* __restrict__ r) {
  __shared__ float tile[32][33];
  const int b = blockIdx.z, t0 = blockIdx.x * 32, f0 = blockIdx.y * 32;
  const int tx = threadIdx.x, ty = threadIdx.y;            // (32,8)
#pragma unroll
  for (int i = 0; i < 4; ++i) {
    int t = t0 + ty + i * 8;
    const float* gp = x + ((size_t)b * T_ + t) * F_ + f0 + tx;
    async_ld32(gp, &tile[ty + i * 8][tx]);
  }
  async_wait_all();
  __syncthreads();
#pragma unroll
  for (int i = 0; i < 4; ++i) {
    int f = f0 + ty + i * 8;
    r[((size_t)b * F_ + f) * T_ + t0 + tx] = tile[tx][ty + i * 8];
  }
}

// ---------------- van Herk / Gil-Werman sliding windows ----------------
template <int OP>   // 0 = max, 1 = min, 2 = add
__device__ __forceinline__ float opf(float a, float b) {
  if (OP == 0) return fmaxf(a, b);
  if (OP == 1) return fminf(a, b);
  return a + b;
}

// per-segment prefix (Rb) and suffix (Lb) reductions; thread s owns segment s
template <int OP>
__device__ __forceinline__ void build_lr(const float* __restrict__ src,
                                         float* __restrict__ Lb, float* __restrict__ Rb,
                                         int tid) {
  for (int s = tid; s < NSEG; s += 256) {
    int s0 = s * SSEG;
    int s1 = s0 + SSEG - 1; if (s1 > T_ - 1) s1 = T_ - 1;
    float acc = src[s0]; Rb[s0] = acc;
    for (int i = s0 + 1; i <= s1; ++i) { acc = opf<OP>(acc, src[i]); Rb[i] = acc; }
    acc = src[s1]; Lb[s1] = acc;
    for (int i = s1 - 1; i >= s0; --i) { acc = opf<OP>(acc, src[i]); Lb[i] = acc; }
  }
}

// clamped window [i-HW_, i+HW_]: spans <=2 segments; clamping only ever hits one side,
// so the same-segment case always has lo==segstart (use Rb) or hi==segend (use Lb)
template <int OP>
__device__ __forceinline__ float query_win(const float* __restrict__ Lb,
                                           const float* __restrict__ Rb, int i) {
  int lo = i - HW_; if (lo < 0) lo = 0;
  int hi = i + HW_; if (hi > T_ - 1) hi = T_ - 1;
  int sl = lo / SSEG, sh = hi / SSEG;
  if (sl != sh) return opf<OP>(Lb[lo], Rb[hi]);
  return (lo == sl * SSEG) ? Rb[hi] : Lb[lo];
}

// ---------------- EMD sift: one column per block, LDS-resident ----------------
__global__ __launch_bounds__(256) void k_sift(float* __restrict__ r, float* __restrict__ hcol,
                                              const int* __restrict__ target, int k) {
  __shared__ float h[T_];
  __shared__ float w[T_];
  __shared__ float Lb[T_];
  __shared__ float Rb[T_];
  const int col = blockIdx.x;          // b*F + f
  const int f = col & (F_ - 1);
  const size_t base = (size_t)col * T_;
  const int tid = threadIdx.x;
  float ev[T_ / 256];                  // envelope, registers

#pragma unroll
  for (int j = 0; j < T_ / (256 * 4); ++j) {   // column load, 16B per lane
    int i4 = tid + j * 256;
    async_ld128(r + base + (size_t)i4 * 4, &h[i4 * 4]);
  }
  async_wait_all();
  __syncthreads();

  for (int s = 0; s < NS_; ++s) {
    // upper envelope: moving-avg of windowed max
    build_lr<0>(h, Lb, Rb, tid);
    __syncthreads();
#pragma unroll
    for (int j = 0; j < T_ / 256; ++j) { int i = tid + j * 256; w[i] = query_win<0>(Lb, Rb, i); }
    __syncthreads();
    build_lr<2>(w, Lb, Rb, tid);
    __syncthreads();
#pragma unroll
    for (int j = 0; j < T_ / 256; ++j) {
      int i = tid + j * 256;
      int lo = i - HW_; if (lo < 0) lo = 0;
      int hi = i + HW_; if (hi > T_ - 1) hi = T_ - 1;
      ev[j] = 0.5f * query_win<2>(Lb, Rb, i) / (float)(hi - lo + 1);
    }
    __syncthreads();
    // lower envelope: moving-avg of windowed min
    build_lr<1>(h, Lb, Rb, tid);
    __syncthreads();
#pragma unroll
    for (int j = 0; j < T_ / 256; ++j) { int i = tid + j * 256; w[i] = query_win<1>(Lb, Rb, i); }
    __syncthreads();
    build_lr<2>(w, Lb, Rb, tid);
    __syncthreads();
#pragma unroll
    for (int j = 0; j < T_ / 256; ++j) {
      int i = tid + j * 256;
      int lo = i - HW_; if (lo < 0) lo = 0;
      int hi = i + HW_; if (hi > T_ - 1) hi = T_ - 1;
      ev[j] += 0.5f * query_win<2>(Lb, Rb, i) / (float)(hi - lo + 1);
    }
    __syncthreads();
#pragma unroll
    for (int j = 0; j < T_ / 256; ++j) { int i = tid + j * 256; h[i] -= ev[j]; }
    __syncthreads();
  }

  const float keep = (k < target[f]) ? 1.f : 0.f;
#pragma unroll
  for (int j = 0; j < T_ / 256; ++j) {
    int i = tid + j * 256;
    float hv = h[i];
    hcol[base + i] = hv * keep;   // masked IMF for outputs/Hilbert
    r[base + i] -= hv;            // residual update uses UNMASKED IMF
  }
}

// ---------------- WMMA radix-16 FFT (4096 = 16^3), table-driven twiddles ----------------
__device__ __forceinline__ int drev12(int i) {   // base-16 digit reversal
  return ((i & 15) << 8) | (i & 240) | ((i >> 8) & 15);
}

__device__ void fft3_wmma(float* re, float* im,
                          const float* c256, const float* s256,
                          const float* c16r, const float* s16r, float sgn) {
  const int lane = threadIdx.x & 31;
  const int wv = threadIdx.x >> 5;     // 8 waves
  const int m = lane & 15;
  const int hf = lane >> 4;
  for (int stage = 0; stage < 3; ++stage) {
    for (int rep = 0; rep < 2; ++rep) {
      const int g = wv + rep * 8;      // matrix id 0..15, disjoint index sets
      int off, rs, cs;
      if (stage == 0)      { off = 16 * g;  rs = 256; cs = 1;  }   // DFT over n2
      else if (stage == 1) { off = 256 * g; rs = 16;  cs = 1;  }   // DFT over n1
      else                 { off = 256 * g; rs = 1;   cs = 16; }   // DFT over n0
      v8f yr = {0.f,0.f,0.f,0.f,0.f,0.f,0.f,0.f};
      v8f yi = {0.f,0.f,0.f,0.f,0.f,0.f,0.f,0.f};
#pragma unroll
      for (int q = 0; q < 4; ++q) {
        const int ka = 4 * q + 2 * hf;
        // A = DFT16: W[m][k] = cos(2pi mk/16) + i*sgn*sin ; reuse w256 table (period 16)
        const int e0 = ((m * ka) & 15) << 4;
        const int e1 = ((m * (ka + 1)) & 15) << 4;
        float c0 = c256[e0],       c1 = c256[e1];
        float s0 = sgn * s256[e0], s1 = sgn * s256[e1];
        v2f Ar  = {c0, c1};
        v2f Ai  = {s0, s1};
        v2f AiN = {-s0, -s1};
        const int ib0 = off + ka * rs + m * cs;
        const int ib1 = ib0 + rs;
        v2f Br = {re[ib0], re[ib1]};
        v2f Bi = {im[ib0], im[ib1]};
        yr = WMMA4(Ar,  Br, yr);
        yr = WMMA4(AiN, Bi, yr);
        yi = WMMA4(Ar,  Bi, yi);
        yi = WMMA4(Ai,  Br, yi);
      }
      // twiddle + in-place store (C/D layout: M = v + 8*hf, N = lane&15)
#pragma unroll
      for (int v = 0; v < 8; ++v) {
        const int M = v + 8 * hf;
        const int N = m;
        float ca = 1.f, sa = 0.f;
        if (stage == 0) {                 // w256^(g*M), g*M < 256
          const int e = g * M;
          ca = c256[e]; sa = sgn * s256[e];
        } else if (stage == 1) {          // w4096^(N*(g+16M)) = w256^(e>>4) * w4096^(e&15)
          const int e = N * (g + 16 * M); // < 4096
          float cb = c256[e >> 4], sb = s256[e >> 4];
          float cr = c16r[e & 15], sr = s16r[e & 15];
          ca = cb * cr - sb * sr;
          sa = sgn * (sb * cr + cb * sr);
        }
        float vr = yr[v], vi = yi[v];
        const int di = off + M * rs + N * cs;
        re[di] = vr * ca - vi * sa;
        im[di] = vr * sa + vi * ca;
      }
    }
    __syncthreads();
  }
}

__device__ void permute_drev(float* re, float* im) {
  const int tid = threadIdx.x;
#pragma unroll
  for (int j = 0; j < T_ / 256; ++j) {
    int i = tid + j * 256;
    int d = drev12(i);
    if (d > i) {                      // involution: owner of smaller index swaps
      float a = re[i]; re[i] = re[d]; re[d] = a;
      float b = im[i]; im[i] = im[d]; im[d] = b;
    }
  }
  __syncthreads();
}

__global__ __launch_bounds__(256) void k_hilbert(const float* __restrict__ hcol,
                                                 float* __restrict__ ampcol,
                                                 float* __restrict__ frqcol) {
  __shared__ float re[T_], im[T_];
  __shared__ float c256[256], s256[256];
  __shared__ float c16r[16], s16r[16];
  const int tid = threadIdx.x;
  const size_t base = (size_t)blockIdx.x * T_;

  // twiddle tables (only transcendentals in the kernel)
  {
    float a = TWO_PI_F * (float)tid * (1.f / 256.f);
    c256[tid] = __cosf(a);
    s256[tid] = __sinf(a);
    if (tid < 16) {
      float b = TWO_PI_F * (float)tid * (1.f / 4096.f);
      c16r[tid] = __cosf(b);
      s16r[tid] = __sinf(b);
    }
  }
#pragma unroll
  for (int j = 0; j < T_ / (256 * 4); ++j) {       // async 16B column load into re
    int i4 = tid + j * 256;
    async_ld128(hcol + base + (size_t)i4 * 4, &re[i4 * 4]);
  }
#pragma unroll
  for (int j = 0; j < T_ / 256; ++j) im[tid + j * 256] = 0.f;
  async_wait_all();
  __syncthreads();

  fft3_wmma(re, im, c256, s256, c16r, s16r, -1.f);  // forward DFT (digit-reversed out)
  permute_drev(re, im);                             // natural-order spectrum

#pragma unroll
  for (int j = 0; j < T_ / 256; ++j) {              // analytic-signal filter
    int i = tid + j * 256;
    float gmul = (i == 0 || i == T_ / 2) ? 1.f : (i < T_ / 2 ? 2.f : 0.f);
    re[i] *= gmul; im[i] *= gmul;
  }
  __syncthreads();

  fft3_wmma(re, im, c256, s256, c16r, s16r, 1.f);   // inverse DFT (unnormalized)
  permute_drev(re, im);                             // natural-order time

  const float invN = 1.f / (float)T_;
#pragma unroll
  for (int j = 0; j < T_ / 256; ++j) {
    int i = tid + j * 256;
    float rr = re[i], ii = im[i];
    ampcol[base + i] = sqrtf(rr * rr + ii * ii) * invN;
    re[i] = atan2f(ii, rr);           // stash angle (scale-invariant)
  }
  __syncthreads();
  // inst_freq = diff(unwrap(angle))/2pi == wrapped-diff/2pi (no scan needed)
#pragma unroll
  for (int j = 0; j < T_ / 256; ++j) {
    int i = tid + j * 256;
    float fr = 0.f;
    if (i > 0) {
      float dd = re[i] - re[i - 1];
      float ddmod = fmodf(dd + PI_F, TWO_PI_F);
      if (ddmod < 0.f) ddmod += TWO_PI_F;
      ddmod -= PI_F;
      if (ddmod == -PI_F && dd > 0.f) ddmod = PI_F;
      fr = (fabsf(dd) < PI_F ? dd : ddmod) * (1.f / TWO_PI_F);
    }
    frqcol[base + i] = fr;
  }
}

// ---------------- repack [B,F,T] -> [B,k,T,F]: pure async LDS-DMA transpose ----------------
__global__ __launch_bounds__(256) void k_repack(const float* __restrict__ hcol,
                                                const float* __restrict__ ampcol,
                                                const float* __restrict__ frqcol,
                                                float* __restrict__ imf_out,
                                                float* __restrict__ amp_out,
                                                float* __restrict__ frq_out,
                                                size_t bstride) {
  __shared__ float tile[32][33];
  const int b = blockIdx.z, t0 = blockIdx.x * 32, f0 = blockIdx.y * 32;
  const int tx = threadIdx.x, ty = threadIdx.y;
  const float* srcs[3] = {hcol, ampcol, frqcol};
  float* dsts[3] = {imf_out, amp_out, frq_out};
  for (int a = 0; a < 3; ++a) {
    async_wait_all();           // previous iteration's stores done before tile reuse
    __syncthreads();
#pragma unroll
    for (int i = 0; i < 4; ++i) {
      int f = f0 + ty + i * 8;
      async_ld32(srcs[a] + ((size_t)b * F_ + f) * T_ + t0 + tx, &tile[ty + i * 8][tx]);
    }
    async_wait_all();
    __syncthreads();
#pragma unroll
    for (int i = 0; i < 4; ++i) {
      int t = t0 + ty + i * 8;
      async_st32(dsts[a] + (size_t)b * bstride + (size_t)t * F_ + f0 + tx,
                 &tile[tx][ty + i * 8]);
    }
  }
  async_wait_all();
}

// ---------------- elementwise / reduction tail ----------------
__global__ void k_residue(const float* __restrict__ x, const float* __restrict__ imfs,
                          float* __restrict__ res) {
  size_t i = (size_t)blockIdx.x * 256 + threadIdx.x;
  if (i >= (size_t)B_ * T_ * F_) return;
  size_t b = i / ((size_t)T_ * F_);
  size_t rem = i % ((size_t)T_ * F_);
  const float* ib = imfs + b * (size_t)K_ * T_ * F_ + rem;
  float s = 0.f;
#pragma unroll
  for (int k = 0; k < K_; ++k) s += ib[(size_t)k * T_ * F_];
  res[i] = x[i] - s;
}

__global__ __launch_bounds__(128) void k_energy_part(const float* __restrict__ x,
                                                     const float* __restrict__ imfs,
                                                     const float* __restrict__ res,
                                                     float* __restrict__ orig_p,
                                                     float* __restrict__ dec_p) {
  const int b = blockIdx.x >> 5, c = blockIdx.x & 31;   // 32 chunks of 128 t
  const int f = threadIdx.x;
  float oe = 0.f, de = 0.f;
  for (int tt = 0; tt < T_ / 32; ++tt) {
    int t = c * (T_ / 32) + tt;
    size_t base = ((size_t)b * T_ + t) * F_ + f;
    float xv = x[base]; oe += xv * xv;
    float rv = res[base]; de += rv * rv;
    size_t ib = ((size_t)b * K_ * T_ + t) * F_ + f;
#pragma unroll
    for (int k = 0; k < K_; ++k) { float v = imfs[ib + (size_t)k * T_ * F_]; de += v * v; }
  }
  orig_p[(size_t)blockIdx.x * F_ + f] = oe;
  dec_p[(size_t)blockIdx.x * F_ + f] = de;
}

__global__ void k_energy_final(const float* __restrict__ orig_p, const float* __restrict__ dec_p,
                               float* __restrict__ ratio) {
  int i = blockIdx.x * 256 + threadIdx.x;
  if (i >= B_ * F_) return;
  int b = i / F_, f = i % F_;
  float oe = 0.f, de = 0.f;
  for (int c = 0; c < 32; ++c) {
    oe += orig_p[((size_t)b * 32 + c) * F_ + f];
    de += dec_p[((size_t)b * 32 + c) * F_ + f];
  }
  ratio[i] = de > 0.f ? sqrtf(oe / fmaxf(de, 1e-12f)) : 1.f;
}

__global__ __launch_bounds__(128) void k_dom_part(const float* __restrict__ ifrq,
                                                  const float* __restrict__ iamp,
                                                  float* __restrict__ domn_p,
                                                  float* __restrict__ domd_p) {
  const int bk = blockIdx.x >> 5, c = blockIdx.x & 31;
  const int f = threadIdx.x;
  float nf = 0.f, na = 0.f;
  for (int tt = 0; tt < T_ / 32; ++tt) {
    int t = c * (T_ / 32) + tt;
    size_t base = ((size_t)bk * T_ + t) * F_ + f;
    float a = iamp[base];
    nf += ifrq[base] * a;
    na += a;
  }
  domn_p[(size_t)blockIdx.x * F_ + f] = nf;
  domd_p[(size_t)blockIdx.x * F_ + f] = na;
}

__global__ void k_rank(const float* __restrict__ domn_p, const float* __restrict__ domd_p,
                       const int* __restrict__ boundary, float* __restrict__ hmask) {
  int i = blockIdx.x * 256 + threadIdx.x;
  if (i >= B_ * F_) return;
  int b = i / F_, f = i % F_;
  float dom[K_];
#pragma unroll
  for (int k = 0; k < K_; ++k) {
    float num = 0.f, den = 0.f;
    for (int c = 0; c < 32; ++c) {
      num += domn_p[(((size_t)(b * K_ + k)) * 32 + c) * F_ + f];
      den += domd_p[(((size_t)(b * K_ + k)) * 32 + c) * F_ + f];
    }
    dom[k] = den > 0.f ? num / fmaxf(den, 1e-12f) : 0.f;
  }
  int bnd = boundary[f];
#pragma unroll
  for (int k = 0; k < K_; ++k) {
    int rk = 0;
#pragma unroll
    for (int j = 0; j < K_; ++j)
      rk += (dom[j] > dom[k]) || (dom[j] == dom[k] && j < k);
    hmask[((size_t)(b * K_ + k)) * F_ + f] = (rk < bnd) ? 1.f : 0.f;
  }
}

__global__ void k_scale(float* __restrict__ p, const float* __restrict__ ratio,
                        size_t n, size_t bstride) {
  size_t i = (size_t)blockIdx.x * 256 + threadIdx.x;
  if (i >= n) return;
  int b = (int)(i / bstride);
  int f = (int)(i % F_);
  p[i] *= ratio[b * F_ + f];
}

__global__ void k_highlow(const float* __restrict__ imfs, const float* __restrict__ imp,
                          const float* __restrict__ hmask,
                          float* __restrict__ hf, float* __restrict__ lf) {
  size_t i = (size_t)blockIdx.x * 256 + threadIdx.x;
  if (i >= (size_t)B_ * T_ * F_) return;
  size_t b = i / ((size_t)T_ * F_);
  size_t rem = i % ((size_t)T_ * F_);
  int f = (int)(i % F_);
  float hs = 0.f, ls = 0.f;
#pragma unroll
  for (int k = 0; k < K_; ++k) {
    float w = imfs[b * (size_t)K_ * T_ * F_ + (size_t)k * T_ * F_ + rem] * imp[k];
    float mk = hmask[((size_t)(b * K_ + k)) * F_ + f];
    hs += w * mk;
    ls += w * (1.f - mk);
  }
  hf[i] = hs;
  lf[i] = ls;
}

// ---------------- launcher ----------------
extern "C" void kernel_launch(void* const* d_in, const int* in_sizes, int n_in,
                              void* d_out, int out_size, void* d_ws, size_t ws_size,
                              hipStream_t stream) {
  (void)in_sizes; (void)n_in; (void)out_size; (void)ws_size;
  const float* x   = (const float*)d_in[0];
  const float* cs  = (const float*)d_in[1];
  const float* imp = (const float*)d_in[2];
  const float* fb  = (const float*)d_in[3];

  const size_t TF  = (size_t)T_ * F_;     // 524288
  const size_t KTF = (size_t)K_ * TF;     // 2621440

  float* out    = (float*)d_out;
  float* o_imfs = out;                    // [B,K,T,F]
  float* o_res  = o_imfs + (size_t)B_ * KTF;
  float* o_hi   = o_res  + (size_t)B_ * TF;
  float* o_lo   = o_hi   + (size_t)B_ * TF;
  float* o_ifrq = o_lo   + (size_t)B_ * TF;   // [B,K,T,F]
  float* o_iamp = o_ifrq + (size_t)B_ * KTF;  // [B,K,T,F]

  float* ws     = (float*)d_ws;
  float* r      = ws;                          // [B,F,T] residual
  float* hcol   = r      + (size_t)B_ * TF;    // [B,F,T] masked IMF
  float* ampcol = hcol   + (size_t)B_ * TF;
  float* frqcol = ampcol + (size_t)B_ * TF;
  float* orig_p = frqcol + (size_t)B_ * TF;    // [B,32,F]
  float* dec_p  = orig_p + (size_t)B_ * 32 * F_;
  float* domn_p = dec_p  + (size_t)B_ * 32 * F_;   // [B*K,32,F]
  float* domd_p = domn_p + (size_t)B_ * K_ * 32 * F_;
  float* ratio  = domd_p + (size_t)B_ * K_ * 32 * F_;  // [B,F]
  float* hmask  = ratio  + (size_t)B_ * F_;            // [B,K,F]
  int*   target = (int*)(hmask + (size_t)B_ * K_ * F_);
  int*   bnd    = target + F_;

  dim3 tb(32, 8);
  dim3 tg(T_ / 32, F_ / 32, B_);

  k_params<<<1, 128, 0, stream>>>(cs, fb, target, bnd);
  k_transpose<<<tg, tb, 0, stream>>>(x, r);

  for (int k = 0; k < K_; ++k) {
    k_sift<<<B_ * F_, 256, 0, stream>>>(r, hcol, target, k);
    k_hilbert<<<B_ * F_, 256, 0, stream>>>(hcol, ampcol, frqcol);
    k_repack<<<tg, tb, 0, stream>>>(hcol, ampcol, frqcol,
                                    o_imfs + (size_t)k * TF,
                                    o_iamp + (size_t)k * TF,
                                    o_ifrq + (size_t)k * TF, KTF);
  }

  k_residue<<<(int)((B_ * TF) / 256), 256, 0, stream>>>(x, o_imfs, o_res);
  k_energy_part<<<B_ * 32, 128, 0, stream>>>(x, o_imfs, o_res, orig_p, dec_p);
  k_energy_final<<<4, 256, 0, stream>>>(orig_p, dec_p, ratio);
  k_dom_part<<<B_ * K_ * 32, 128, 0, stream>>>(o_ifrq, o_iamp, domn_p, domd_p); // pre-scale; ratio cancels
  k_rank<<<4, 256, 0, stream>>>(domn_p, domd_p, bnd, hmask);
  k_scale<<<(int)((B_ * KTF) / 256), 256, 0, stream>>>(o_imfs, ratio, (size_t)B_ * KTF, KTF);
  k_scale<<<(int)((B_ * KTF) / 256), 256, 0, stream>>>(o_iamp, ratio, (size_t)B_ * KTF, KTF);
  k_scale<<<(int)((B_ * TF) / 256), 256, 0, stream>>>(o_res, ratio, (size_t)B_ * TF, TF);
  k_highlow<<<(int)((B_ * TF) / 256), 256, 0, stream>>>(o_imfs, imp, hmask, o_hi, o_lo);
}